// AdaptiveComputationTime_55448027792024
// MI455X (gfx1250) — compile-verified
//
#include <hip/hip_runtime.h>
#include <math.h>

typedef float v2f __attribute__((ext_vector_type(2)));
typedef float v4f __attribute__((ext_vector_type(4)));
typedef float v8f __attribute__((ext_vector_type(8)));

constexpr int B = 8, M = 4096, H = 512;
constexpr int N = B * M;               // 32768 tokens
constexpr float THRESHOLD = 0.99f;

// ---------------------------------------------------------------------------
// Kernel 1: per-batch inclusive scan of run -> rank = clip(cumsum(run)-1, 0)
// 1024 threads (32 wave32), 4 elements/thread, Hillis-Steele block scan.
// ---------------------------------------------------------------------------
__global__ __launch_bounds__(1024)
void act_scan_rank(const int* __restrict__ run, int* __restrict__ rank) {
  __shared__ int sdata[1024];
  const int b   = blockIdx.x;
  const int tid = threadIdx.x;
  const int base = b * M + tid * 4;
  const int v0 = run[base + 0] != 0;
  const int v1 = run[base + 1] != 0;
  const int v2 = run[base + 2] != 0;
  const int v3 = run[base + 3] != 0;
  const int s0 = v0, s1 = s0 + v1, s2 = s1 + v2, s3 = s2 + v3;
  sdata[tid] = s3;
  __syncthreads();
  for (int off = 1; off < 1024; off <<= 1) {
    const int t = (tid >= off) ? sdata[tid - off] : 0;
    __syncthreads();
    sdata[tid] += t;
    __syncthreads();
  }
  const int excl = sdata[tid] - s3;
  int r0 = excl + s0 - 1; if (r0 < 0) r0 = 0;
  int r1 = excl + s1 - 1; if (r1 < 0) r1 = 0;
  int r2 = excl + s2 - 1; if (r2 < 0) r2 = 0;
  int r3 = excl + s3 - 1; if (r3 < 0) r3 = 0;
  rank[base + 0] = r0;
  rank[base + 1] = r1;
  rank[base + 2] = r2;
  rank[base + 3] = r3;
}

// ---------------------------------------------------------------------------
// Kernel 3: per-batch scan of run_new -> pos[j] = m (scatter), S[b] = count
// ---------------------------------------------------------------------------
__global__ __launch_bounds__(1024)
void act_scan_pos(const int* __restrict__ run_new, int* __restrict__ pos,
                  int* __restrict__ S) {
  __shared__ int sdata[1024];
  const int b   = blockIdx.x;
  const int tid = threadIdx.x;
  const int base = b * M + tid * 4;
  const int v0 = run_new[base + 0] != 0;
  const int v1 = run_new[base + 1] != 0;
  const int v2 = run_new[base + 2] != 0;
  const int v3 = run_new[base + 3] != 0;
  const int s0 = v0, s1 = s0 + v1, s2 = s1 + v2, s3 = s2 + v3;
  sdata[tid] = s3;
  __syncthreads();
  for (int off = 1; off < 1024; off <<= 1) {
    const int t = (tid >= off) ? sdata[tid - off] : 0;
    __syncthreads();
    sdata[tid] += t;
    __syncthreads();
  }
  const int excl = sdata[tid] - s3;
  const int m0 = tid * 4;
  if (v0) pos[b * M + excl + s0 - 1] = m0 + 0;
  if (v1) pos[b * M + excl + s1 - 1] = m0 + 1;
  if (v2) pos[b * M + excl + s2 - 1] = m0 + 2;
  if (v3) pos[b * M + excl + s3 - 1] = m0 + 3;
  if (tid == 1023) S[b] = sdata[1023];
}

// ---------------------------------------------------------------------------
// Kernel 2: main ACT step. 256 threads = 8 wave32, 16 tokens per wave,
// 128 tokens per block, 256 blocks.
// Phase 1: WMMA f32 16x16x4 dot(h_row, W) accumulation over K=512.
//          B fragment columns all replicated with the same W chunk so every
//          D column equals the desired dot -> layout-robust.
// Phase 2a: per-token halting math, write acc/rem, run_new.
// Phase 2b: coalesced float4 streaming update of weighted_h.
// Cache policy: h/W regular-temporal (h is re-read: keep in 192MB L2);
// all streaming read-once/write-once traffic is non-temporal.
// ---------------------------------------------------------------------------
__global__ __launch_bounds__(256)
void act_main(const float* __restrict__ h, const int* __restrict__ run,
              const float* __restrict__ acc_p, const float* __restrict__ wh,
              const float* __restrict__ rem, const float* __restrict__ W,
              const float* __restrict__ bias, const int* __restrict__ rank,
              int* __restrict__ run_new, float* __restrict__ out_wh,
              float* __restrict__ out_acc, float* __restrict__ out_rem) {
  __shared__ float wlds[H];        // W staged in LDS
  __shared__ float dots[128];      // one dot per token
  __shared__ float peff_s[128];
  __shared__ int   src_s[128];     // gathered global row index, -1 if !run

  const int tid   = threadIdx.x;
  const int blk   = blockIdx.x;            // 0..255
  const int b     = blk >> 5;              // 32 chunks per batch
  const int m0    = (blk & 31) * 128;      // first token of this block

  if (tid < 128) {
    *(v4f*)&wlds[tid * 4] = *(const v4f*)&W[tid * 4];
  }
  __syncthreads();

  // ---- Phase 1: WMMA dots (EXEC all-ones here: no divergence) ----
  const int wave = tid >> 5;
  const int lane = tid & 31;
  const int tl   = lane & 15;              // token within wave (A-row M)
  const int half = lane >> 4;              // K pair select: 0 -> K0,K1 ; 1 -> K2,K3
  {
    const int g  = b * M + m0 + wave * 16 + tl;
    const int rk = rank[g];                // clipped rank: always a valid row
    const float* rowp = h + ((size_t)(b * M) + (size_t)rk) * H + half * 2;
    v8f c = {0.f, 0.f, 0.f, 0.f, 0.f, 0.f, 0.f, 0.f};
    for (int k = 0; k < H / 4; ++k) {
      const v2f a  = *(const v2f*)(rowp + 4 * k);
      const v2f bb = *(const v2f*)&wlds[4 * k + 2 * half];
      c = __builtin_amdgcn_wmma_f32_16x16x4_f32(false, a, false, bb,
                                                (short)0, c, false, false);
    }
    // C layout: VGPR v holds M=v (lanes 0-15) / M=v+8 (lanes 16-31); all N equal.
    if (lane == 0) {
      #pragma unroll
      for (int v = 0; v < 8; ++v) dots[wave * 16 + v] = c[v];
    } else if (lane == 16) {
      #pragma unroll
      for (int v = 0; v < 8; ++v) dots[wave * 16 + 8 + v] = c[v];
    }
  }
  __syncthreads();

  // ---- Phase 2a: per-token scalar ACT math ----
  if (tid < 128) {
    const int g  = b * M + m0 + tid;
    const int rv = run[g] != 0;
    const float runf = rv ? 1.f : 0.f;
    float p = 1.f / (1.f + expf(-(dots[tid] + bias[0])));
    p *= runf;                                  // zero at exited slots
    const float accn = __builtin_nontemporal_load(&acc_p[g]) + p;
    const float mc = ((accn < THRESHOLD) && rv) ? 1.f : 0.f;
    const float me = runf - mc;                 // exit mask
    const float pe = p * mc + (1.f - (accn - p)) * me;
    __builtin_nontemporal_store(accn, &out_acc[g]);
    const float rv0 = __builtin_nontemporal_load(&rem[g]);
    __builtin_nontemporal_store(rv0 + pe * me, &out_rem[g]);
    run_new[g]  = (mc != 0.f) ? 1 : 0;
    peff_s[tid] = pe;
    src_s[tid]  = rv ? (b * M + rank[g]) : -1;
  }
  __syncthreads();

  // ---- Phase 2b: weighted_h_new = h_full*pe + wh*(1-pe), float4 coalesced ----
  const int c0 = tid & 127;
  const int i  = c0 * 4;
  for (int r = tid >> 7; r < 128; r += 2) {
    const float pe  = peff_s[r];
    const int   src = src_s[r];
    const int   gRow = b * M + m0 + r;
    if (r + 2 < 128) {
      const int nsrc = src_s[r + 2];
      if (nsrc >= 0) __builtin_prefetch(h + (size_t)nsrc * H + i, 0, 0);
    }
    v4f hv = {0.f, 0.f, 0.f, 0.f};
    if (src >= 0) hv = *(const v4f*)(h + (size_t)src * H + i);     // RT: L2-resident
    const v4f wv = __builtin_nontemporal_load((const v4f*)(wh + (size_t)gRow * H + i));
    const v4f o  = hv * pe + wv * (1.f - pe);
    __builtin_nontemporal_store(o, (v4f*)(out_wh + (size_t)gRow * H + i));
  }
}

// ---------------------------------------------------------------------------
// Kernel 4: h_out = pack(h_full, run_new). One block (128 thr) per output row.
// Fully covers d_out region (zeros the tail rows). h read is L2-hot (RT),
// h_out store is non-temporal streaming.
// ---------------------------------------------------------------------------
__global__ __launch_bounds__(128)
void act_pack(const float* __restrict__ h, const int* __restrict__ rank,
              const int* __restrict__ pos, const int* __restrict__ S,
              float* __restrict__ h_out) {
  const int j   = blockIdx.x;       // global output row, 0..N-1
  const int b   = j / M;
  const int jj  = j - b * M;
  const int tid = threadIdx.x;
  v4f val = {0.f, 0.f, 0.f, 0.f};
  if (jj < S[b]) {
    const int m   = pos[j];
    const int src = b * M + rank[b * M + m];   // run_new implies run
    val = *(const v4f*)(h + (size_t)src * H + tid * 4);
  }
  __builtin_nontemporal_store(val, (v4f*)(h_out + (size_t)j * H + tid * 4));
}

// ---------------------------------------------------------------------------
extern "C" void kernel_launch(void* const* d_in, const int* in_sizes, int n_in,
                              void* d_out, int out_size, void* d_ws, size_t ws_size,
                              hipStream_t stream) {
  const float* h     = (const float*)d_in[0];
  const int*   run   = (const int*)d_in[1];
  const float* acc_p = (const float*)d_in[2];
  const float* wh    = (const float*)d_in[3];
  const float* rem   = (const float*)d_in[4];
  const float* W     = (const float*)d_in[5];
  const float* bias  = (const float*)d_in[6];

  float* out     = (float*)d_out;
  float* h_out   = out;                              // [B,M,H]
  float* out_wh  = out + (size_t)N * H;              // [B,M,H]
  float* out_acc = out + 2 * (size_t)N * H;          // [B,M]
  float* out_rem = out_acc + N;                      // [B,M]

  int* rank    = (int*)d_ws;        // N ints
  int* run_new = rank + N;          // N ints
  int* pos     = run_new + N;       // N ints
  int* S       = pos + N;           // B ints

  act_scan_rank<<<B, 1024, 0, stream>>>(run, rank);
  act_main<<<N / 128, 256, 0, stream>>>(h, run, acc_p, wh, rem, W, bias, rank,
                                        run_new, out_wh, out_acc, out_rem);
  act_scan_pos<<<B, 1024, 0, stream>>>(run_new, pos, S);
  act_pack<<<N, 128, 0, stream>>>(h, rank, pos, S, h_out);
}